// Densenet_with_skip_53154515255362
// MI455X (gfx1250) — compile-verified
//
#include <hip/hip_runtime.h>

// ---------------------------------------------------------------------------
// Fused GaussianBlur(5x5) -> Sobel -> 45-degree angle binning -> 2-tap NMS ->
// ReLU gate, single pass over HBM. Memory-bound: ~67MB total traffic
// (~2.9us roofline at 23.3 TB/s); compute ~35 VALU ops/px sits under that.
// CDNA5 path: GLOBAL_LOAD_ASYNC_TO_LDS_B32 staging + s_wait_asynccnt.
// ---------------------------------------------------------------------------

#define HW     512
#define TILE   32
#define SA_N   38          // reconst tile: TILE + 2*3 halo
#define SA_W   40          // padded LDS row stride (bank-conflict avoidance)
#define TH_N   34          // blurred width needed: TILE + 2*1 halo
#define TH_W   36
#define SB_W   36

typedef __attribute__((address_space(3))) int* lds_ip;
typedef __attribute__((address_space(1))) int* gbl_ip;

__device__ __forceinline__ void async_copy_b32(const float* gp, float* lp) {
#if __has_builtin(__builtin_amdgcn_global_load_async_to_lds_b32)
  __builtin_amdgcn_global_load_async_to_lds_b32(
      (gbl_ip)const_cast<float*>(gp), (lds_ip)lp, 0, 0);
#else
  unsigned lds_off = (unsigned)(unsigned long long)(lds_ip)lp;
  asm volatile("global_load_async_to_lds_b32 %0, %1, off"
               :: "v"(lds_off), "v"((unsigned long long)gp)
               : "memory");
#endif
}

__device__ __forceinline__ void wait_async0() {
#if __has_builtin(__builtin_amdgcn_s_wait_asynccnt)
  __builtin_amdgcn_s_wait_asynccnt(0);
#else
  asm volatile("s_wait_asynccnt 0" ::: "memory");
#endif
}

__global__ __launch_bounds__(256)
void canny_nms_fused(const float* __restrict__ reconst,
                     const float* __restrict__ gk,   // [1,1,5,5] gaussian
                     const float* __restrict__ nk,   // [8,1,3,3] nms kernels
                     float* __restrict__ out)
{
  __shared__ float sA[SA_N * SA_W];   // reconst tile, zero-padded at image edge
  __shared__ float sT[SA_N * TH_W];   // horizontally blurred (38 rows x 34 cols)
  __shared__ float sB[TH_N * SB_W];   // blurred (34 x 34), zero outside image

  const int tid = threadIdx.x;
  const int bx = blockIdx.x, by = blockIdx.y, n = blockIdx.z;
  const int gx0 = bx * TILE - 3, gy0 = by * TILE - 3;
  const float* img = reconst + (size_t)n * (HW * HW);

  // ---- stage reconst tile into LDS via async copies (one writer per cell) --
  for (int idx = tid; idx < SA_N * SA_N; idx += 256) {
    int r = idx / SA_N;
    int c = idx - r * SA_N;
    int gy = gy0 + r, gx = gx0 + c;
    float* lp = &sA[r * SA_W + c];
    if ((unsigned)gy < HW && (unsigned)gx < HW) {
      async_copy_b32(img + gy * HW + gx, lp);
    } else {
      *lp = 0.0f;              // zero padding (nn.Conv2d semantics)
    }
  }

  // separable gaussian weights: k = outer(u,u)  =>  u[j] = sqrt(k[j][j]).
  // (overlaps with in-flight async loads; uniform scalar-cacheable reads)
  float u0 = sqrtf(gk[0]);
  float u1 = sqrtf(gk[6]);
  float u2 = sqrtf(gk[12]);
  float u3 = sqrtf(gk[18]);
  float u4 = sqrtf(gk[24]);
  const float c0 = nk[4];      // NMS center tap  (=  1.0)
  const float vv = nk[5];      // NMS neighbor tap (= -0.5)

  wait_async0();
  __syncthreads();

  // ---- horizontal blur: 38 rows x 34 cols -------------------------------
  for (int idx = tid; idx < SA_N * TH_N; idx += 256) {
    int r = idx / TH_N, c = idx - r * TH_N;
    const float* p = &sA[r * SA_W + c];
    sT[r * TH_W + c] = u0 * p[0] + u1 * p[1] + u2 * p[2] + u3 * p[3] + u4 * p[4];
  }
  __syncthreads();

  // ---- vertical blur: 34 x 34; force zero outside image ------------------
  // (halo cells outside the image must read as 0 for the zero-padded NMS conv)
  for (int idx = tid; idx < TH_N * TH_N; idx += 256) {
    int r = idx / TH_N, c = idx - r * TH_N;
    int gy = by * TILE - 1 + r, gx = bx * TILE - 1 + c;
    float val = 0.0f;
    if ((unsigned)gy < HW && (unsigned)gx < HW) {
      const float* p = &sT[r * TH_W + c];
      val = u0 * p[0] + u1 * p[TH_W] + u2 * p[2 * TH_W]
          + u3 * p[3 * TH_W] + u4 * p[4 * TH_W];
    }
    sB[r * SB_W + c] = val;
  }
  __syncthreads();

  // ---- sobel + angle bin + 2-tap NMS + relu gate, 4 px / thread ----------
  const int ybase = by * TILE - 1, xbase = bx * TILE - 1;
  float* outp = out + (size_t)n * (HW * HW);
  const int lx = tid & 31;
  const int X  = bx * TILE + lx;
  const int xm1 = max(X - 1, 0) - xbase;        // replicate pad for sobel
  const int xp1 = min(X + 1, HW - 1) - xbase;
  const int x0  = lx + 1;

#pragma unroll
  for (int k = 0; k < 4; ++k) {
    int ly = (tid >> 5) + (k << 3);
    int Y  = by * TILE + ly;
    int ym1 = max(Y - 1, 0) - ybase;
    int yp1 = min(Y + 1, HW - 1) - ybase;
    int y0  = ly + 1;

    float tl = sB[ym1 * SB_W + xm1], tc = sB[ym1 * SB_W + x0], tr = sB[ym1 * SB_W + xp1];
    float ml = sB[y0  * SB_W + xm1],                            mr = sB[y0  * SB_W + xp1];
    float bl = sB[yp1 * SB_W + xm1], bc = sB[yp1 * SB_W + x0], br = sB[yp1 * SB_W + xp1];

    float gxv = (tr + 2.0f * mr + br) - (tl + 2.0f * ml + bl);
    float gyv = (bl + 2.0f * bc + br) - (tl + 2.0f * tc + tr);

    // round(atan2/45) mod 8 without atan2: octant by tan(22.5deg) boundaries.
    // d and d+4 use negated offsets, so only the offset class (d&3) matters;
    // the <= choices reproduce jnp.round's half-to-even at the exact boundaries.
    const float T = 0.4142135623730951f;        // tan(22.5 deg)
    float ax = fabsf(gxv), ay = fabsf(gyv);
    int dy, dx;
    if (ay <= T * ax)                             { dy = 0; dx = 1; }   // 0 / 180
    else if (ax <= T * ay)                        { dy = 1; dx = 0; }   // +-90
    else if ((gxv >= 0.0f) == (gyv >= 0.0f))      { dy = 1; dx = 1; }   // 45 / -135
    else                                          { dy = 1; dx = -1; }  // 135 / -45

    // min(b + v*n1, b + v*n2) with v<0  ==  b + v*max(n1,n2)
    float b0 = sB[y0 * SB_W + x0];
    float n1 = sB[(y0 + dy) * SB_W + (x0 + dx)];
    float n2 = sB[(y0 - dy) * SB_W + (x0 - dx)];
    float m  = c0 * b0 + vv * fmaxf(n1, n2);

    float rv = sA[(ly + 3) * SA_W + (lx + 3)];  // original reconst value
    outp[Y * HW + X] = rv * fmaxf(m, 0.0f);
  }
}

extern "C" void kernel_launch(void* const* d_in, const int* in_sizes, int n_in,
                              void* d_out, int out_size, void* d_ws, size_t ws_size,
                              hipStream_t stream) {
  (void)n_in; (void)out_size; (void)d_ws; (void)ws_size;
  const float* reconst = (const float*)d_in[0];
  const float* gk      = (const float*)d_in[1];
  const float* nk      = (const float*)d_in[2];
  float* out           = (float*)d_out;
  int batch = in_sizes[0] / (HW * HW);          // 32
  dim3 grid(HW / TILE, HW / TILE, batch);       // 16 x 16 x 32
  canny_nms_fused<<<grid, dim3(256), 0, stream>>>(reconst, gk, nk, out);
}